// MetricNN_9251359556270
// MI455X (gfx1250) — compile-verified
//
#include <hip/hip_runtime.h>
#include <math.h>

// Problem constants (match the reference)
#define N   8192
#define K1  10      // layer-1 inner dim
#define M1  3       // layer-1 output dim == stage-2 inner dim
#define TJ  512     // columns per block tile
#define TI  32      // rows per block tile

typedef __attribute__((ext_vector_type(4))) float v4f;

__device__ __forceinline__ float max3f(float a, float b, float c) {
    // Guarantee the single-instruction 3-way max on gfx1250.
    float d;
    asm("v_max3_num_f32 %0, %1, %2, %3" : "=v"(d) : "v"(a), "v"(b), "v"(c));
    return d;
}

// ---------------------------------------------------------------------------
// Stage 1: out[n,m] = max_k(|x[n,k]| + |w1[m,k]|)   (8192 x 3, 96 KB total)
// One thread per row n. w1 is wave-uniform -> scalar loads, stays in SGPRs.
// ---------------------------------------------------------------------------
__global__ __launch_bounds__(256) void talayer1_kernel(
    const float* __restrict__ x, const float* __restrict__ w,
    float* __restrict__ out)
{
    const int n = blockIdx.x * 256 + threadIdx.x;

    float wa[M1][K1];
#pragma unroll
    for (int m = 0; m < M1; ++m)
#pragma unroll
        for (int k = 0; k < K1; ++k)
            wa[m][k] = fabsf(w[m * K1 + k]);   // uniform -> s_load

    float xa[K1];
#pragma unroll
    for (int k = 0; k < K1; ++k)
        xa[k] = fabsf(x[(size_t)n * K1 + k]);

#pragma unroll
    for (int m = 0; m < M1; ++m) {
        float best = xa[0] + wa[m][0];
#pragma unroll
        for (int k = 1; k < K1; ++k)
            best = fmaxf(best, xa[k] + wa[m][k]);
        out[n * M1 + m] = best;
    }
}

// ---------------------------------------------------------------------------
// Stage 2: dist[i,j] = max_{k<3}(a[i,k] + a[j,k])  -- store-bandwidth bound.
//
// Block = 256 threads (8 wave32), tile = TI x TJ = 32 x 512 (64 KB of output).
//  - Each thread owns 4 consecutive columns: 12 floats via 3 coalesced b128
//    loads, kept in VGPRs for the whole tile.
//  - Row triples are WAVE-UNIFORM (ty = tid>>7 is constant per wave32):
//    readfirstlane makes that provable, so the 16 row reads per wave become
//    scalar/broadcast loads off the VALU critical path (no LDS, no barrier).
//  - Per row: 12 v_add_f32 (VOPD-pairable) + 4 v_max3_num_f32, then one
//    16B-aligned NON-TEMPORAL global_store_b128 (write-once output).
// ---------------------------------------------------------------------------
__global__ __launch_bounds__(256) void tropical_dist_kernel(
    const float* __restrict__ a, float* __restrict__ dist)
{
    const int tid = threadIdx.x;
    const int jt0 = blockIdx.x * TJ;
    const int it0 = blockIdx.y * TI;

    const int tx = tid & 127;                                  // column-quad owner
    const int ty = __builtin_amdgcn_readfirstlane(tid >> 7);   // wave-uniform 0/1

    const int j = jt0 + tx * 4;

    // 12 contiguous floats = components 0..2 of columns j..j+3 (16B aligned:
    // j % 4 == 0 -> j*3 floats is a multiple of 48 bytes).
    const v4f* ap = (const v4f*)(a + (size_t)j * M1);
    const v4f c0 = ap[0];
    const v4f c1 = ap[1];
    const v4f c2 = ap[2];

    // Unpack AoS -> per-component registers for the 4 owned columns.
    const float aj0[4] = {c0[0], c0[3], c1[2], c2[1]};
    const float aj1[4] = {c0[1], c1[0], c1[3], c2[2]};
    const float aj2[4] = {c0[2], c1[1], c2[0], c2[3]};

    // Uniform base of this wave's row triples: rows il = ty + 2r.
    const float* __restrict__ arow = a + (size_t)(it0 + ty) * M1;

#pragma unroll
    for (int r = 0; r < TI / 2; ++r) {
        const int il = ty + 2 * r;                 // local row 0..31
        const float ai0 = arow[6 * r + 0];         // uniform -> scalar/broadcast
        const float ai1 = arow[6 * r + 1];
        const float ai2 = arow[6 * r + 2];

        v4f o;
#pragma unroll
        for (int q = 0; q < 4; ++q)
            o[q] = max3f(ai0 + aj0[q], ai1 + aj1[q], ai2 + aj2[q]);

        // Write-once 256 MB result: non-temporal b128 store, fully coalesced
        // (128 threads x 16 B = 2 KB contiguous per row group).
        __builtin_nontemporal_store(
            o, (v4f*)(dist + (size_t)(it0 + il) * N + j));
    }
}

extern "C" void kernel_launch(void* const* d_in, const int* in_sizes, int n_in,
                              void* d_out, int out_size, void* d_ws, size_t ws_size,
                              hipStream_t stream) {
    const float* x  = (const float*)d_in[0];   // (8192, 10) fp32
    const float* w1 = (const float*)d_in[1];   // (3, 10)    fp32
    float* dist = (float*)d_out;               // (8192, 8192) fp32
    float* a    = (float*)d_ws;                // stage-1 result: 8192*3 floats (96 KB)

    talayer1_kernel<<<N / 256, 256, 0, stream>>>(x, w1, a);

    dim3 grid(N / TJ, N / TI);                 // 16 x 256 = 4096 blocks
    tropical_dist_kernel<<<grid, 256, 0, stream>>>(a, dist);
}